// PathDDPM_91250875171016
// MI455X (gfx1250) — compile-verified
//
#include <hip/hip_runtime.h>
#include <math.h>

// ---------------------------------------------------------------------------
// DDPM q(x_t | x_0) forward noising, MI455X (gfx1250) — single fused kernel.
//
// Roofline: 256 MiB HBM traffic vs ~33 MFLOP => pure bandwidth problem,
// floor ~11.5 us at 23.3 TB/s. The whole DDPM schedule is a compile-time
// constant, so it's baked into __constant__ memory (4 KB) and fetched with
// one SGPR-indexed s_load per block. Runtime work per lane is just:
//   8 x B128 non-temporal loads, 32 FMAs, 8 x B128 non-temporal stores.
//   - non-temporal: pure stream, zero reuse — don't rinse the 192 MB L2
//   - wave32: 256-thread blocks = 8 waves; one block per sample
//   - t[b], ac[t_b] block-uniform -> scalarized (readfirstlane -> s_load)
// ---------------------------------------------------------------------------

typedef __attribute__((ext_vector_type(4))) float v4f;

#define DDPM_TIMESTEPS  1000
#define DDPM_BETA_START 1e-4
#define DDPM_BETA_END   0.02

// alphas_cumprod[i] = prod_{j<=i} (1 - beta_j), computed at COMPILE TIME with
// exactly the reference's float32 sequential-cumprod semantics.
struct DdpmTab { float ac[DDPM_TIMESTEPS]; };

constexpr DdpmTab ddpm_make_tab() {
    DdpmTab tab{};
    const double step =
        (DDPM_BETA_END - DDPM_BETA_START) / (double)(DDPM_TIMESTEPS - 1);
    float prod = 1.0f;
    for (int i = 0; i < DDPM_TIMESTEPS; ++i) {
        const float beta = (float)(DDPM_BETA_START + step * (double)i);
        prod = prod * (1.0f - beta);            // f32 cumprod, like jnp
        tab.ac[i] = prod;
    }
    return tab;
}

__constant__ constexpr DdpmTab g_ddpm_tab = ddpm_make_tab();

// One block per sample b: L*2 = 4096 floats = 1024 v4f = 256 threads x 4.
__global__ void __launch_bounds__(256)
ddpm_qsample_kernel(const v4f* __restrict__ x0,
                    const v4f* __restrict__ nz,
                    const int* __restrict__ t,
                    v4f* __restrict__ xt_out,
                    v4f* __restrict__ nz_out,
                    unsigned vec4_per_sample) {
    const unsigned b = blockIdx.x;

    // Block-uniform scalars: t[b] and the schedule entry come back via the
    // scalar path (s_load + SGPR-indexed s_load from constant memory).
    const int   tb = __builtin_amdgcn_readfirstlane(t[b]);
    const float ac = g_ddpm_tab.ac[tb];
    const float a  = sqrtf(ac);                 // sqrt(alphas_cumprod[t])
    const float s  = sqrtf(1.0f - ac);          // sqrt(1 - alphas_cumprod[t])

    const unsigned base = b * vec4_per_sample + threadIdx.x;

    if (vec4_per_sample == 1024u) {
        #pragma unroll
        for (int it = 0; it < 4; ++it) {
            const unsigned idx = base + (unsigned)it * 256u;
            const v4f x = __builtin_nontemporal_load(&x0[idx]);
            const v4f n = __builtin_nontemporal_load(&nz[idx]);
            const v4f y = x * a + n * s;        // contracts to v_fma per lane
            __builtin_nontemporal_store(y, &xt_out[idx]);
            __builtin_nontemporal_store(n, &nz_out[idx]);
        }
    } else {
        // Generic fallback for other shapes.
        for (unsigned i = threadIdx.x; i < vec4_per_sample; i += 256u) {
            const unsigned idx = b * vec4_per_sample + i;
            const v4f x = __builtin_nontemporal_load(&x0[idx]);
            const v4f n = __builtin_nontemporal_load(&nz[idx]);
            const v4f y = x * a + n * s;
            __builtin_nontemporal_store(y, &xt_out[idx]);
            __builtin_nontemporal_store(n, &nz_out[idx]);
        }
    }
}

// ---------------------------------------------------------------------------
// Launch. Inputs (reference setup_inputs order):
//   d_in[0] = x_0   float32 [B, L, 2]
//   d_in[1] = t     int32   [B]
//   d_in[2] = noise float32 [B, L, 2]
// Output: (x_t, noise) concatenated flat in d_out. No workspace needed.
// ---------------------------------------------------------------------------
extern "C" void kernel_launch(void* const* d_in, const int* in_sizes, int n_in,
                              void* d_out, int out_size, void* d_ws, size_t ws_size,
                              hipStream_t stream) {
    (void)n_in; (void)out_size; (void)d_ws; (void)ws_size;

    const v4f* x0 = (const v4f*)d_in[0];
    const int* t  = (const int*)d_in[1];
    const v4f* nz = (const v4f*)d_in[2];

    const int n_elems = in_sizes[0];      // B * L * 2 = 16,777,216 floats
    const int B       = in_sizes[1];      // 4096
    const unsigned vec4_per_sample = (unsigned)((n_elems / B) / 4);  // 1024

    float* outf   = (float*)d_out;
    v4f*   xt_out = (v4f*)outf;
    v4f*   nz_out = (v4f*)(outf + n_elems);

    ddpm_qsample_kernel<<<B, 256, 0, stream>>>(x0, nz, t,
                                               xt_out, nz_out,
                                               vec4_per_sample);
}